// Block_58574763983799
// MI455X (gfx1250) — compile-verified
//
#include <hip/hip_runtime.h>

// Transformer block for MI455X (gfx1250), wave32, fp32 WMMA 16x16x4.
// B=2048 T=64 C=384 H=8 D=48 F=1536.  Memory-bound problem (~480 GFLOP vs
// ~2 GB minimal HBM traffic) -> fuse everything, fp32 WMMA for accuracy,
// async global->LDS (ASYNCcnt) for all tile staging.
//
// Workspace layout (floats), requires ~611 MB:
//   WqkvT [1152][384] | WpT [384][384] | W1T [1536][384] | W2T [384][1536] | QKV [131072][1152]
// Attention writes its output into the Q slots of QKV (cols 0..383).
// x2 (post-attention residual) lives in d_out and is consumed/overwritten by the FFN kernel.

#define BTOK  131072   // B*T rows
#define CDIM  384
#define NHEAD 8
#define HDIM  48
#define TSEQ  64
#define FDIM  1536

typedef float v2f __attribute__((ext_vector_type(2)));
typedef float v8f __attribute__((ext_vector_type(8)));

__device__ __forceinline__ v8f wmma4(v2f a, v2f b, v8f c) {
  // D = A(16x4,f32) * B(4x16,f32) + C(16x16,f32)
  return __builtin_amdgcn_wmma_f32_16x16x4_f32(false, a, false, b, (short)0, c,
                                               false, false);
}

__device__ __forceinline__ float wsum(float v) {
  for (int o = 16; o > 0; o >>= 1) v += __shfl_xor(v, o, 32);
  return v;
}
__device__ __forceinline__ float wmax(float v) {
  for (int o = 16; o > 0; o >>= 1) v = fmaxf(v, __shfl_xor(v, o, 32));
  return v;
}

// Async DMA: 16 bytes global -> LDS, tracked by ASYNCcnt (no VGPR data path).
// GVS addressing: uniform 64-bit SGPR base + per-lane 32-bit byte offset;
// per-lane LDS byte address (low 32 bits of the flat shared pointer).
__device__ __forceinline__ void async_b128(float* lds_dst,
                                           const float* gbase, int byte_off) {
  unsigned lds = (unsigned)(size_t)lds_dst;
  asm volatile("global_load_async_to_lds_b128 %0, %1, %2"
               :
               : "v"(lds), "v"(byte_off), "s"(gbase)
               : "memory");
}
__device__ __forceinline__ void async_wait0() {
  asm volatile("s_wait_asynccnt 0" ::: "memory");
}

// ---------------------------------------------------------------- weight pack
__global__ __launch_bounds__(256) void pack_weights(
    const float* __restrict__ Wq, const float* __restrict__ Wk,
    const float* __restrict__ Wv, const float* __restrict__ Wp,
    const float* __restrict__ W1, const float* __restrict__ W2,
    float* __restrict__ WqkvT, float* __restrict__ WpT,
    float* __restrict__ W1T, float* __restrict__ W2T) {
  int i = blockIdx.x * 256 + threadIdx.x;
  if (i < 1152 * 384) {                       // WqkvT[j][c]
    int j = i / 384, c = i % 384;
    const float* W = (j < 384) ? Wq : (j < 768) ? Wk : Wv;
    int jj = j % 384, h = jj / HDIM, d = jj % HDIM;
    WqkvT[i] = W[(size_t)h * CDIM * HDIM + (size_t)c * HDIM + d];
    return;
  }
  i -= 1152 * 384;
  if (i < 384 * 384) {                        // WpT[n][c] = Wp[c][n]
    int n = i / 384, c = i % 384;
    WpT[i] = Wp[(size_t)c * 384 + n];
    return;
  }
  i -= 384 * 384;
  if (i < 1536 * 384) {                       // W1T[n][k] = W1[k][n]
    int n = i / 384, k = i % 384;
    W1T[i] = W1[(size_t)k * 1536 + n];
    return;
  }
  i -= 1536 * 384;
  if (i < 384 * 1536) {                       // W2T[n][k] = W2[k][n]
    int n = i / 1536, k = i % 1536;
    W2T[i] = W2[(size_t)k * 384 + n];
  }
}

// ------------------------------------------- GEMM (K=384), optional LN / epi
// Block = 256 thr (8 waves). Tile 32(M) x 128(N). A staged in LDS pitch 388
// (16B-aligned rows, conflict-free b64 A-fragment reads). B from packed
// transposed weights (L2 resident).
template <bool FUSE_LN, bool BIAS_RES>
__global__ __launch_bounds__(256) void gemm384(
    const float* __restrict__ A, int lda, const float* __restrict__ Bt,
    const float* __restrict__ gamma, const float* __restrict__ beta,
    const float* __restrict__ bias, const float* __restrict__ res, int ldr,
    float* __restrict__ Cout, int ldc) {
  __shared__ float As[32 * 388];
  const int tid = threadIdx.x, wave = tid >> 5, lane = tid & 31;
  const int l15 = lane & 15, kr = (lane >> 4) * 2;
  const int m0 = blockIdx.x * 32, n0 = blockIdx.y * 128;

  // async-stage A tile 32x384 straight into LDS
  const float* Abase = A + (size_t)m0 * lda;
  for (int i = tid; i < 32 * 96; i += 256) {
    int r = i / 96, c4 = (i % 96) * 4;
    async_b128(&As[r * 388 + c4], Abase, (r * lda + c4) * 4);
  }
  async_wait0();
  __syncthreads();

  if (FUSE_LN) {  // per-row layernorm in LDS (4 rows per wave)
    for (int rr = 0; rr < 4; ++rr) {
      int r = wave * 4 + rr;
      float s = 0.f;
      for (int c = lane; c < 384; c += 32) s += As[r * 388 + c];
      float mu = wsum(s) * (1.f / 384.f);
      float s2 = 0.f;
      for (int c = lane; c < 384; c += 32) {
        float d = As[r * 388 + c] - mu;
        s2 += d * d;
      }
      float rstd = rsqrtf(wsum(s2) * (1.f / 384.f) + 1e-5f);
      for (int c = lane; c < 384; c += 32)
        As[r * 388 + c] = (As[r * 388 + c] - mu) * rstd * gamma[c] + beta[c];
    }
    __syncthreads();
  }

  const int ncol = n0 + wave * 16 + l15;
  const float* Bp = Bt + (size_t)ncol * 384;
  const float* a0 = &As[l15 * 388];
  const float* a1 = &As[(l15 + 16) * 388];
  v8f acc0 = {}, acc1 = {};
  for (int k = 0; k < 384; k += 4) {
    v2f b = *(const v2f*)(Bp + k + kr);
    v2f x0 = *(const v2f*)(a0 + k + kr);
    v2f x1 = *(const v2f*)(a1 + k + kr);
    acc0 = wmma4(x0, b, acc0);
    acc1 = wmma4(x1, b, acc1);
  }

  const int rb = m0 + 8 * (lane >> 4);
  float bv = BIAS_RES ? bias[ncol] : 0.f;
  for (int j = 0; j < 8; ++j) {
    int r0 = rb + j, r1 = rb + 16 + j;
    float o0 = acc0[j], o1 = acc1[j];
    if (BIAS_RES) {
      o0 += bv + res[(size_t)r0 * ldr + ncol];
      o1 += bv + res[(size_t)r1 * ldr + ncol];
    }
    Cout[(size_t)r0 * ldc + ncol] = o0;
    Cout[(size_t)r1 * ldc + ncol] = o1;
  }
}

// ------------------------------------------------------- attention per (b,h)
__global__ __launch_bounds__(256) void attn_kernel(float* __restrict__ QKV) {
  __shared__ float Qs[64 * 52], Ks[64 * 52], Vs[64 * 52], Ps[64 * 68];
  const int b = blockIdx.x, h = blockIdx.y;
  const int tid = threadIdx.x, wave = tid >> 5, lane = tid & 31;
  const int l15 = lane & 15, kr = (lane >> 4) * 2;
  float* base = QKV + (size_t)(b * TSEQ) * 1152 + h * HDIM;

  for (int i = tid; i < 64 * 12; i += 256) {   // async-stage q,k,v (64x48)
    int r = i / 12, c4 = (i % 12) * 4;
    int off = (r * 1152 + c4) * 4;
    async_b128(&Qs[r * 52 + c4], base, off);
    async_b128(&Ks[r * 52 + c4], base, off + 384 * 4);
    async_b128(&Vs[r * 52 + c4], base, off + 768 * 4);
  }
  async_wait0();
  __syncthreads();

  // S = scale * Q K^T, causal-masked, into Ps (pitch 68)
  for (int t = wave; t < 16; t += 8) {
    int mt = (t >> 2) * 16, nt = (t & 3) * 16;
    const float* qp = &Qs[(mt + l15) * 52];
    const float* kp = &Ks[(nt + l15) * 52];
    v8f acc = {};
    for (int k = 0; k < 48; k += 4) {
      v2f a = *(const v2f*)(qp + k + kr);
      v2f bb = *(const v2f*)(kp + k + kr);
      acc = wmma4(a, bb, acc);
    }
    int col = nt + l15, rb = mt + 8 * (lane >> 4);
    for (int j = 0; j < 8; ++j) {
      int row = rb + j;
      float sv = acc[j] * 2.4494897428f;  // HEAD_SIZE * C^-0.5 = sqrt(6)
      Ps[row * 68 + col] = (col <= row) ? sv : -__builtin_inff();
    }
  }
  __syncthreads();

  // causal softmax, 8 rows per wave, lane-parallel over 64 cols
  for (int rr = 0; rr < 8; ++rr) {
    int row = wave * 8 + rr;
    float x0 = Ps[row * 68 + lane], x1 = Ps[row * 68 + 32 + lane];
    float m = wmax(fmaxf(x0, x1));
    float e0 = __expf(x0 - m), e1 = __expf(x1 - m);
    float inv = 1.f / wsum(e0 + e1);
    Ps[row * 68 + lane] = e0 * inv;
    Ps[row * 68 + 32 + lane] = e1 * inv;
  }
  __syncthreads();

  // O = P @ V (64x48), written back into the Q slots of QKV
  for (int t = wave; t < 12; t += 8) {
    int mt = (t / 3) * 16, nt = (t % 3) * 16;
    const float* pp = &Ps[(mt + l15) * 68];
    v8f acc = {};
    for (int k = 0; k < 64; k += 4) {
      v2f a = *(const v2f*)(pp + k + kr);
      v2f bb;
      bb.x = Vs[(k + kr) * 52 + nt + l15];
      bb.y = Vs[(k + kr + 1) * 52 + nt + l15];
      acc = wmma4(a, bb, acc);
    }
    int col = nt + l15, rb = mt + 8 * (lane >> 4);
    for (int j = 0; j < 8; ++j)
      base[(size_t)(rb + j) * 1152 + col] = acc[j];
  }
}

// ------------------------ LN2 + FFN1(ReLU) + FFN2 + bias + residual, fused
// Block = 16 rows. Hidden 16x1536 kept in LDS, processed in 2 chunks of 768.
__global__ __launch_bounds__(256) void ffn_kernel(
    float* __restrict__ X2, const float* __restrict__ g2,
    const float* __restrict__ be2, const float* __restrict__ W1T,
    const float* __restrict__ b1, const float* __restrict__ W2T,
    const float* __restrict__ b2) {
  __shared__ float Xs[16 * 384];   // raw x2 tile (LN stats + residual)
  __shared__ float Hs[16 * 388];   // normalized (A-fragment pitch)
  __shared__ float Us[16 * 772];   // ReLU hidden chunk (768 cols)
  const int tid = threadIdx.x, wave = tid >> 5, lane = tid & 31;
  const int l15 = lane & 15, kr = (lane >> 4) * 2;
  const int m0 = blockIdx.x * 16;

  const float* Xbase = X2 + (size_t)m0 * 384;
  for (int i = tid; i < 16 * 96; i += 256) {
    int r = i / 96, c4 = (i % 96) * 4;
    async_b128(&Xs[r * 384 + c4], Xbase, (r * 384 + c4) * 4);
  }
  async_wait0();
  __syncthreads();

  for (int rr = 0; rr < 2; ++rr) {  // LN2: 2 rows per wave
    int r = wave * 2 + rr;
    float s = 0.f;
    for (int c = lane; c < 384; c += 32) s += Xs[r * 384 + c];
    float mu = wsum(s) * (1.f / 384.f);
    float s2 = 0.f;
    for (int c = lane; c < 384; c += 32) {
      float d = Xs[r * 384 + c] - mu;
      s2 += d * d;
    }
    float rstd = rsqrtf(wsum(s2) * (1.f / 384.f) + 1e-5f);
    for (int c = lane; c < 384; c += 32)
      Hs[r * 388 + c] = (Xs[r * 384 + c] - mu) * rstd * g2[c] + be2[c];
  }
  __syncthreads();

  v8f outv[3];
  for (int i = 0; i < 3; ++i) outv[i] = (v8f){};

  for (int chunk = 0; chunk < 2; ++chunk) {
    int nbase = chunk * 768;
    // stage 1: Us = relu(Hs @ W1 + b1), 48 col-tiles, 6 per wave
    for (int tt = wave; tt < 48; tt += 8) {
      int nc = tt * 16;
      const float* Bp = W1T + (size_t)(nbase + nc + l15) * 384;
      const float* Ap = &Hs[l15 * 388];
      v8f acc = {};
      for (int k = 0; k < 384; k += 4) {
        v2f a = *(const v2f*)(Ap + k + kr);
        v2f bb = *(const v2f*)(Bp + k + kr);
        acc = wmma4(a, bb, acc);
      }
      int col = nc + l15, rb = 8 * (lane >> 4);
      float bv = b1[nbase + col];
      for (int j = 0; j < 8; ++j)
        Us[(rb + j) * 772 + col] = fmaxf(acc[j] + bv, 0.f);
    }
    __syncthreads();
    // stage 2: out += Us @ W2[chunk rows]
    for (int ot = 0; ot < 3; ++ot) {
      int ncol = wave * 16 + ot * 128 + l15;
      const float* Bp = W2T + (size_t)ncol * 1536 + nbase;
      const float* Ap = &Us[l15 * 772];
      v8f acc = outv[ot];
      for (int k = 0; k < 768; k += 4) {
        v2f a = *(const v2f*)(Ap + k + kr);
        v2f bb = *(const v2f*)(Bp + k + kr);
        acc = wmma4(a, bb, acc);
      }
      outv[ot] = acc;
    }
    __syncthreads();
  }

  const int rb = 8 * (lane >> 4);
  for (int ot = 0; ot < 3; ++ot) {
    int col = wave * 16 + ot * 128 + l15;
    float bv = b2[col];
    for (int j = 0; j < 8; ++j) {
      int r = rb + j;
      X2[(size_t)(m0 + r) * 384 + col] = outv[ot][j] + bv + Xs[r * 384 + col];
    }
  }
}

// ----------------------------------------------------------------- launcher
extern "C" void kernel_launch(void* const* d_in, const int* in_sizes, int n_in,
                              void* d_out, int out_size, void* d_ws,
                              size_t ws_size, hipStream_t stream) {
  const float* x     = (const float*)d_in[0];
  const float* ln1_g = (const float*)d_in[1];
  const float* ln1_b = (const float*)d_in[2];
  const float* ln2_g = (const float*)d_in[3];
  const float* ln2_b = (const float*)d_in[4];
  const float* Wq    = (const float*)d_in[5];
  const float* Wk    = (const float*)d_in[6];
  const float* Wv    = (const float*)d_in[7];
  const float* Wp    = (const float*)d_in[8];
  const float* bp    = (const float*)d_in[9];
  const float* W1    = (const float*)d_in[10];
  const float* b1    = (const float*)d_in[11];
  const float* W2    = (const float*)d_in[12];
  const float* b2    = (const float*)d_in[13];
  float* out = (float*)d_out;

  float* ws = (float*)d_ws;
  float* WqkvT = ws;                       // 1152*384
  float* WpT   = WqkvT + 1152 * 384;       // 384*384
  float* W1T   = WpT + 384 * 384;          // 1536*384
  float* W2T   = W1T + 1536 * 384;         // 384*1536
  float* QKV   = W2T + 384 * 1536;         // BTOK*1152

  // 1. pack weights into transposed WMMA-friendly layouts
  pack_weights<<<6912, 256, 0, stream>>>(Wq, Wk, Wv, Wp, W1, W2, WqkvT, WpT,
                                         W1T, W2T);
  // 2. LN1 + QKV projection: QKV[BT,1152]
  gemm384<true, false><<<dim3(BTOK / 32, 1152 / 128), 256, 0, stream>>>(
      x, CDIM, WqkvT, ln1_g, ln1_b, nullptr, nullptr, 0, QKV, 1152);
  // 3. attention per (b,h); O overwrites the Q slots of QKV
  attn_kernel<<<dim3(BTOK / TSEQ, NHEAD), 256, 0, stream>>>(QKV);
  // 4. output projection + bias + residual -> x2 in d_out
  gemm384<false, true><<<dim3(BTOK / 32, 384 / 128), 256, 0, stream>>>(
      QKV, 1152, WpT, nullptr, nullptr, bp, x, CDIM, out, CDIM);
  // 5. LN2 + MLP + residual, in-place on d_out
  ffn_kernel<<<BTOK / 16, 256, 0, stream>>>(out, ln2_g, ln2_b, W1T, b1, W2T,
                                            b2);
}